// Classifier_18433999634718
// MI455X (gfx1250) — compile-verified
//
#include <hip/hip_runtime.h>
#include <math.h>

typedef float v2f __attribute__((ext_vector_type(2)));
typedef float v8f __attribute__((ext_vector_type(8)));

static inline int cdiv(int a, int b){ return (a + b - 1) / b; }
static __device__ __forceinline__ int imin(int a, int b){ return a < b ? a : b; }
static __device__ __forceinline__ int imax(int a, int b){ return a > b ? a : b; }

// ---------------------------------------------------------------------------
// Float atomic max via CAS (logits can be negative; lmax initialized to -inf)
// ---------------------------------------------------------------------------
static __device__ __forceinline__ float atomicMaxF(float* addr, float val){
  int* ia = (int*)addr;
  int old = __float_as_int(*addr);
  while (__int_as_float(old) < val){
    int assumed = old;
    old = atomicCAS(ia, assumed, __float_as_int(val));
    if (old == assumed) break;
  }
  return __int_as_float(old);
}

// ---------------------------------------------------------------------------
// Generic WMMA GEMM, full fp32 precision via V_WMMA_F32_16X16X4_F32.
// C[M,N] = A[M,K] @ B[K,N] (+ bias[N]) (optional ReLU).
// transB==1 : B[k][n] = Bm[n*K + k]   (i.e. C = A @ Bm^T)
//
// Block = 128 threads (4 waves); block tile 64(M) x 64(N); wave tile 16x64.
// OOB handling: M/N indices are CLAMPED (not branched) -- WMMA rows/cols are
// independent, so data loaded for clamped rows/cols feeds only accumulator
// rows/cols that are never stored. Only the K tail (K%4) needs value masking.
// K-loop is unrolled x2 with ping-pong fragment registers (no copies), so the
// WMMA group for one set depends only on its own loads and the other set's
// loads stay in flight -> partial s_wait_loadcnt, real load/WMMA overlap.
// ---------------------------------------------------------------------------
__global__ __launch_bounds__(128)
void wmma_gemm(const float* __restrict__ Am, const float* __restrict__ Bm,
               const float* __restrict__ bias, float* __restrict__ Cm,
               int M, int N, int K, int transB, int relu)
{
  const int lane = threadIdx.x & 31;
  const int wave = threadIdx.x >> 5;
  const int half = lane >> 4;
  const int l15  = lane & 15;
  const int mbase = blockIdx.y * 64 + wave * 16;
  const int nbase = blockIdx.x * 64;
  const int mA  = mbase + l15;
  const int mAc = imin(mA, M - 1);                    // clamped, always in-bounds
  int ncol[4];
  const float* Bbase[4];
  const size_t sK = transB ? 1 : (size_t)N;           // k-stride inside B
#pragma unroll
  for (int t = 0; t < 4; ++t){
    ncol[t] = nbase + t*16 + l15;
    int nc  = imin(ncol[t], N - 1);                   // clamped column
    Bbase[t] = transB ? (Bm + (size_t)nc*K) : (Bm + nc);
  }
  const float* Arow = Am + (size_t)mAc * K;

  v8f acc[4];
#pragma unroll
  for (int t = 0; t < 4; ++t)
#pragma unroll
    for (int i = 0; i < 8; ++i) acc[t][i] = 0.f;

  const int kOff  = half * 2;
  const int Kmain = K & ~3;

  auto loadfrag = [&](int k0, bool guardK, v2f& A_, v2f (&B_)[4]){
    const int kA  = k0 + kOff;
    const int k0c = guardK ? imin(kA,     K - 1) : kA;
    const int k1c = guardK ? imin(kA + 1, K - 1) : (kA + 1);
    A_.x = Arow[k0c];
    A_.y = Arow[k1c];
#pragma unroll
    for (int t = 0; t < 4; ++t){
      float x = Bbase[t][(size_t)k0c * sK];
      float y = Bbase[t][(size_t)k1c * sK];
      if (guardK){
        if (kA     >= K) x = 0.f;                     // zero only the K tail
        if (kA + 1 >= K) y = 0.f;
      }
      B_[t].x = x; B_[t].y = y;
    }
  };
  auto domm = [&](const v2f& A_, const v2f (&B_)[4]){
#pragma unroll
    for (int t = 0; t < 4; ++t)
      acc[t] = __builtin_amdgcn_wmma_f32_16x16x4_f32(false, A_, false, B_[t],
                                                     (short)0, acc[t], false, false);
  };

  v2f a0, b0[4], a1, b1[4];
  if (Kmain > 0){
    const int steps = Kmain >> 2;                     // >= 1
    loadfrag(0, false, a0, b0);
    int i = 0;
    for (; i + 2 <= steps; i += 2){
      loadfrag((i + 1) * 4, false, a1, b1);           // (i+1)*4 <= Kmain-4 here
      __builtin_prefetch(Bbase[0] + (size_t)imin(i*4 + 32, K - 1) * sK);
      domm(a0, b0);
      loadfrag(imin((i + 2) * 4, Kmain - 4), false, a0, b0);  // clamped pipeline load
      domm(a1, b1);
    }
    if (i < steps) domm(a0, b0);                      // odd leftover step
  }
  if (Kmain < K){                                     // K % 4 tail, scalar-predicated
    loadfrag(Kmain, true, a0, b0);
    domm(a0, b0);
  }

#pragma unroll
  for (int t = 0; t < 4; ++t){
    int n = ncol[t];
    if (n >= N) continue;
    float bv = bias ? bias[n] : 0.f;
#pragma unroll
    for (int i = 0; i < 8; ++i){
      int m = mbase + i + half*8;
      if (m < M){
        float v = acc[t][i] + bv;
        if (relu) v = fmaxf(v, 0.f);
        Cm[(size_t)m*N + n] = v;
      }
    }
  }
}

// ---------------------------------------------------------------------------
// Implicit-GEMM 3x3x3 pad-1 stride-1 conv3d, fp32 WMMA.
// Y[oc, n] = sum_k W[oc, k] * im2col(X)[k, n],  K = IC*27,  n = sample*SP + sp.
// im2col column synthesized on the fly; pad positions are CLAMPED to valid
// addresses and zeroed with a select (no exec-divergent loads, no OOB reads).
// Same ping-pong double-buffered K-loop as wmma_gemm.
// ---------------------------------------------------------------------------
__global__ __launch_bounds__(128)
void wmma_conv3d(const float* __restrict__ X, const float* __restrict__ Wt,
                 const float* __restrict__ bias, float* __restrict__ Y,
                 int S, int IC, int OC, int D, int H, int W)
{
  const int lane = threadIdx.x & 31;
  const int wave = threadIdx.x >> 5;
  const int half = lane >> 4;
  const int l15  = lane & 15;
  const int HW   = H * W;
  const int SP   = D * HW;
  const int Ntot = S * SP;
  const int K    = IC * 27;
  const int mbase = blockIdx.y * 64 + wave * 16;
  const int nbase = blockIdx.x * 64;
  const int mA  = mbase + l15;
  const int mAc = imin(mA, OC - 1);
  const float* Wrow = Wt + (size_t)mAc * K;

  int od[4], oh[4], ow[4], nidx[4]; size_t xb[4];
#pragma unroll
  for (int t = 0; t < 4; ++t){
    int n = nbase + t*16 + l15;
    nidx[t] = n;
    int nn = imin(n, Ntot - 1);                       // clamp: OOB cols unused
    int s = nn / SP, sp = nn - s*SP;
    od[t] = sp / HW;
    int r = sp - od[t]*HW;
    oh[t] = r / W; ow[t] = r - oh[t]*W;
    xb[t] = (size_t)s * IC * SP;
  }
  v8f acc[4];
#pragma unroll
  for (int t = 0; t < 4; ++t)
#pragma unroll
    for (int i = 0; i < 8; ++i) acc[t][i] = 0.f;

  const int kOff  = half * 2;
  const int Kmain = K & ~3;

  auto loadfrag = [&](int k0, bool guardK, v2f& A_, v2f (&B_)[4]){
    const int kA  = k0 + kOff;
    const int k0c = guardK ? imin(kA,     K - 1) : kA;
    const int k1c = guardK ? imin(kA + 1, K - 1) : (kA + 1);
    A_.x = Wrow[k0c];
    A_.y = Wrow[k1c];
#pragma unroll
    for (int kk = 0; kk < 2; ++kk){
      const int k   = (kk == 0) ? k0c : k1c;
      const bool kv = !guardK || ((kA + kk) < K);
      const int ic  = k / 27;
      const int t27 = k - ic*27;
      const int kd  = t27 / 9;
      const int r9  = t27 - kd*9;
      const int kh  = r9 / 3;
      const int kw  = r9 - kh*3;
#pragma unroll
      for (int t = 0; t < 4; ++t){
        int id = od[t] - 1 + kd;
        int ih = oh[t] - 1 + kh;
        int iw = ow[t] - 1 + kw;
        bool ok = kv && (unsigned)id < (unsigned)D &&
                        (unsigned)ih < (unsigned)H &&
                        (unsigned)iw < (unsigned)W;
        int idc = imin(imax(id, 0), D - 1);           // v_med3-style clamps
        int ihc = imin(imax(ih, 0), H - 1);
        int iwc = imin(imax(iw, 0), W - 1);
        float v = X[xb[t] + ((size_t)ic*D + idc)*HW + (size_t)ihc*W + iwc];
        v = ok ? v : 0.f;
        if (kk == 0) B_[t].x = v; else B_[t].y = v;
      }
    }
  };
  auto domm = [&](const v2f& A_, const v2f (&B_)[4]){
#pragma unroll
    for (int t = 0; t < 4; ++t)
      acc[t] = __builtin_amdgcn_wmma_f32_16x16x4_f32(false, A_, false, B_[t],
                                                     (short)0, acc[t], false, false);
  };

  v2f a0, b0[4], a1, b1[4];
  if (Kmain > 0){
    const int steps = Kmain >> 2;
    loadfrag(0, false, a0, b0);
    int i = 0;
    for (; i + 2 <= steps; i += 2){
      loadfrag((i + 1) * 4, false, a1, b1);
      domm(a0, b0);
      loadfrag(imin((i + 2) * 4, Kmain - 4), false, a0, b0);
      domm(a1, b1);
    }
    if (i < steps) domm(a0, b0);
  }
  if (Kmain < K){
    loadfrag(Kmain, true, a0, b0);
    domm(a0, b0);
  }

#pragma unroll
  for (int t = 0; t < 4; ++t){
    int n = nidx[t];
    if (n >= Ntot) continue;
    int s = n / SP, sp = n - s*SP;
#pragma unroll
    for (int i = 0; i < 8; ++i){
      int m = mbase + i + half*8;
      if (m < OC)
        Y[((size_t)s*OC + m)*SP + sp] = acc[t][i] + bias[m];
    }
  }
}

// ---------------------------------------------------------------------------
// Elementwise / reduction helper kernels
// ---------------------------------------------------------------------------
__global__ void fill_f32(float* p, float v, int n){
  int i = blockIdx.x*blockDim.x + threadIdx.x;
  if (i < n) p[i] = v;
}

// BatchNorm stats: stats[c]=mean, stats[C+c]=rsqrt(var+eps). NCHW-style layout.
__global__ void bn_stats_nchw(const float* __restrict__ x, float* stats,
                              int S, int C, int SP){
  int c = blockIdx.x;
  int n = S*SP;
  float s = 0.f, s2 = 0.f;
  for (int i = threadIdx.x; i < n; i += 256){
    int smp = i / SP, sp = i - smp*SP;
    float v = x[((size_t)smp*C + c)*SP + sp];
    s += v; s2 += v*v;
  }
  __shared__ float sh[256], sh2[256];
  sh[threadIdx.x] = s; sh2[threadIdx.x] = s2;
  __syncthreads();
  for (int o = 128; o > 0; o >>= 1){
    if ((int)threadIdx.x < o){
      sh[threadIdx.x]  += sh[threadIdx.x+o];
      sh2[threadIdx.x] += sh2[threadIdx.x+o];
    }
    __syncthreads();
  }
  if (threadIdx.x == 0){
    float m  = sh[0] / n;
    float va = fmaxf(sh2[0]/n - m*m, 0.f);
    stats[c]     = m;
    stats[C + c] = rsqrtf(va + 1e-5f);
  }
}

__global__ void bn_apply_nchw(float* x, const float* stats, const float* g,
                              const float* b, int S, int C, int SP, int relu){
  int idx = blockIdx.x*blockDim.x + threadIdx.x;
  int total = S*C*SP;
  if (idx >= total) return;
  int c = (idx / SP) % C;
  float v = (x[idx] - stats[c]) * stats[C + c] * g[c] + b[c];
  if (relu) v = fmaxf(v, 0.f);
  x[idx] = v;
}

// Row-major [R, C] feature-last BN
__global__ void bn_stats_rows(const float* __restrict__ x, float* stats, int R, int C){
  int c = blockIdx.x;
  float s = 0.f, s2 = 0.f;
  for (int i = threadIdx.x; i < R; i += 256){
    float v = x[(size_t)i*C + c];
    s += v; s2 += v*v;
  }
  __shared__ float sh[256], sh2[256];
  sh[threadIdx.x] = s; sh2[threadIdx.x] = s2;
  __syncthreads();
  for (int o = 128; o > 0; o >>= 1){
    if ((int)threadIdx.x < o){
      sh[threadIdx.x]  += sh[threadIdx.x+o];
      sh2[threadIdx.x] += sh2[threadIdx.x+o];
    }
    __syncthreads();
  }
  if (threadIdx.x == 0){
    float m  = sh[0] / R;
    float va = fmaxf(sh2[0]/R - m*m, 0.f);
    stats[c]     = m;
    stats[C + c] = rsqrtf(va + 1e-5f);
  }
}

__global__ void bn_apply_rows(float* x, const float* stats, const float* g,
                              const float* b, int R, int C, int relu){
  int idx = blockIdx.x*blockDim.x + threadIdx.x;
  int total = R*C;
  if (idx >= total) return;
  int c = idx % C;
  float v = (x[idx] - stats[c]) * stats[C + c] * g[c] + b[c];
  if (relu) v = fmaxf(v, 0.f);
  x[idx] = v;
}

// 2x2x2 stride-2 max pool with per-dim pad flags (pad value = -inf).
__global__ void maxpool3d_k(const float* __restrict__ in, float* __restrict__ out,
                            int S, int C, int Di, int Hi, int Wi,
                            int Do, int Ho, int Wo, int pd, int ph, int pw){
  int idx = blockIdx.x*blockDim.x + threadIdx.x;
  int total = S*C*Do*Ho*Wo;
  if (idx >= total) return;
  int ow = idx % Wo; int tt = idx / Wo;
  int oh = tt % Ho;  tt /= Ho;
  int od = tt % Do;  tt /= Do;
  int c  = tt % C;   int s = tt / C;
  float m = -INFINITY;
#pragma unroll
  for (int kd = 0; kd < 2; ++kd)
#pragma unroll
    for (int kh = 0; kh < 2; ++kh)
#pragma unroll
      for (int kw = 0; kw < 2; ++kw){
        int id = od*2 - pd + kd;
        int ih = oh*2 - ph + kh;
        int iw = ow*2 - pw + kw;
        if ((unsigned)id < (unsigned)Di && (unsigned)ih < (unsigned)Hi &&
            (unsigned)iw < (unsigned)Wi)
          m = fmaxf(m, in[(((size_t)s*C + c)*Di + id)*Hi*Wi + (size_t)ih*Wi + iw]);
      }
  out[idx] = m;
}

// GRU pointwise gates (torch order r,z,n). Writes h and h_all[:,t,:].
__global__ void gru_gate(const float* __restrict__ gx, const float* __restrict__ gh,
                         float* __restrict__ h, float* __restrict__ hall,
                         int t, int T, int Bn, int Hd){
  int idx = blockIdx.x*blockDim.x + threadIdx.x;
  if (idx >= Bn*Hd) return;
  int b = idx / Hd, j = idx - b*Hd;
  const float* gxr = gx + (size_t)(b*T + t)*3*Hd;
  const float* ghr = gh + (size_t)b*3*Hd;
  float r = 1.f / (1.f + expf(-(gxr[j]        + ghr[j])));
  float z = 1.f / (1.f + expf(-(gxr[Hd + j]   + ghr[Hd + j])));
  float n = tanhf(gxr[2*Hd + j] + r * ghr[2*Hd + j]);
  float hp = h[idx];
  float hn = (1.f - z)*n + z*hp;
  h[idx] = hn;
  hall[(size_t)(b*T + t)*Hd + j] = hn;
}

// feat[:,0] = node_x ; feat[:,1:513] = enc[cell_index]
__global__ void feat_gather(const float* __restrict__ node_x,
                            const int* __restrict__ cell_index,
                            const float* __restrict__ enc,
                            float* __restrict__ feat, int Nn){
  int idx = blockIdx.x*blockDim.x + threadIdx.x;
  int total = Nn * 513;
  if (idx >= total) return;
  int i = idx / 513, j = idx - i*513;
  feat[idx] = (j == 0) ? node_x[i] : enc[(size_t)cell_index[i]*512 + (j - 1)];
}

__global__ void deg_meanea_accum(const int* __restrict__ dst,
                                 const float* __restrict__ ea,
                                 float* deg, float* meanea, int E){
  int e = blockIdx.x*blockDim.x + threadIdx.x;
  if (e >= E) return;
  int d = dst[e];
  atomicAdd(&deg[d], 1.f);
  atomicAdd(&meanea[2*d],     ea[2*e]);
  atomicAdd(&meanea[2*d + 1], ea[2*e + 1]);
}

__global__ void meanea_norm(float* meanea, const float* deg, int Nn){
  int i = blockIdx.x*blockDim.x + threadIdx.x;
  if (i >= Nn) return;
  float dg = fmaxf(deg[i], 1.f);
  meanea[2*i]     /= dg;
  meanea[2*i + 1] /= dg;
}

// Per-edge (real edges + N self loops) GATv2 attention logits + running max.
__global__ void gat_edge_logits(const float* __restrict__ xl, const float* __restrict__ xr,
                                const int* __restrict__ src, const int* __restrict__ dst,
                                const float* __restrict__ ea, const float* __restrict__ meanea,
                                const float* __restrict__ we, const float* __restrict__ att,
                                float* __restrict__ logit, float* lmax,
                                int E, int Nn, int Hh, int Cc){
  int e = blockIdx.x*blockDim.x + threadIdx.x;
  int E2 = E + Nn;
  if (e >= E2) return;
  int s, d; float a0, a1;
  if (e < E){ s = src[e]; d = dst[e]; a0 = ea[2*e]; a1 = ea[2*e + 1]; }
  else      { s = d = e - E; a0 = meanea[2*s]; a1 = meanea[2*s + 1]; }
  const int HC = Hh * Cc;
  const float* xls = xl + (size_t)s*HC;
  const float* xrd = xr + (size_t)d*HC;
  for (int h = 0; h < Hh; ++h){
    float acc = 0.f;
    for (int c = 0; c < Cc; ++c){
      int j = h*Cc + c;
      float m = xls[j] + xrd[j] + a0*we[j] + a1*we[HC + j];
      m = (m > 0.f) ? m : 0.2f*m;           // LeakyReLU(0.2)
      acc += att[j] * m;
    }
    logit[(size_t)e*Hh + h] = acc;
    atomicMaxF(&lmax[(size_t)d*Hh + h], acc);
  }
}

__global__ void gat_edge_soft(float* __restrict__ elog, const float* __restrict__ lmax,
                              float* wsum, const int* __restrict__ dst,
                              int E, int Nn, int Hh){
  int idx = blockIdx.x*blockDim.x + threadIdx.x;
  int E2 = E + Nn;
  if (idx >= E2*Hh) return;
  int e = idx / Hh, h = idx - e*Hh;
  int d = (e < E) ? dst[e] : (e - E);
  float w = expf(elog[idx] - lmax[(size_t)d*Hh + h]);
  elog[idx] = w;
  atomicAdd(&wsum[(size_t)d*Hh + h], w);
}

__global__ void gat_edge_aggr(const float* __restrict__ elog, const float* __restrict__ wsum,
                              const float* __restrict__ xl,
                              const int* __restrict__ dst, const int* __restrict__ src,
                              float* __restrict__ gout, int E, int Nn, int Hh, int Cc){
  int idx = blockIdx.x*blockDim.x + threadIdx.x;
  const int HC = Hh*Cc;
  int total = (E + Nn) * HC;
  if (idx >= total) return;
  int e = idx / HC, j = idx - e*HC;
  int h = j / Cc;
  int d, s;
  if (e < E){ d = dst[e]; s = src[e]; } else { d = s = e - E; }
  float alpha = elog[(size_t)e*Hh + h] / wsum[(size_t)d*Hh + h];
  atomicAdd(&gout[(size_t)d*HC + j], alpha * xl[(size_t)s*HC + j]);
}

__global__ void gat_finalize(float* gout, const float* deg, const float* bias,
                             int Nn, int HC){
  int idx = blockIdx.x*blockDim.x + threadIdx.x;
  if (idx >= Nn*HC) return;
  int i = idx / HC, j = idx - i*HC;
  gout[idx] = gout[idx] / (deg[i] + 1.f) + bias[j];
}

__global__ void final_out(const float* __restrict__ g3, const unsigned char* __restrict__ mask,
                          const float* __restrict__ y, float* __restrict__ out, int Nn){
  int i = blockIdx.x*blockDim.x + threadIdx.x;
  if (i >= Nn) return;
  float p = 1.f / (1.f + expf(-g3[i]));
  bool m = (mask[i] != 0);
  out[i]      = m ? p    : 0.f;
  out[Nn + i] = m ? y[i] : 0.f;
}

// ---------------------------------------------------------------------------
// Host orchestration
// ---------------------------------------------------------------------------
extern "C" void kernel_launch(void* const* d_in, const int* in_sizes, int n_in,
                              void* d_out, int out_size, void* d_ws, size_t ws_size,
                              hipStream_t stream)
{
  (void)n_in; (void)out_size; (void)ws_size;
  const float* Xb        = (const float*)d_in[0];   // [900,5,5,6,6]
  const float* node_x    = (const float*)d_in[1];
  const int*   cell_idx  = (const int*)  d_in[2];
  const int*   edge_idx  = (const int*)  d_in[3];   // [2,E]
  const float* edge_attr = (const float*)d_in[4];   // [E,2]
  const unsigned char* train_mask = (const unsigned char*)d_in[5];
  const float* yv        = (const float*)d_in[6];
  auto P = [&](int i)->const float* { return (const float*)d_in[i]; };

  const int Nn = in_sizes[1];
  const int E  = in_sizes[3] / 2;
  const int S  = in_sizes[0] / (5*5*6*6);   // 900
  const int T  = 25;
  const int BL = S / T;                     // 36
  const int* src = edge_idx;
  const int* dst = edge_idx + E;

  // ------------- workspace allocator (fp32), two-phase overlap -------------
  float* ws = (float*)d_ws;
  size_t off = 0;
  auto A = [&](size_t n)->float*{ float* p = ws + off; off += (n + 63) & ~(size_t)63; return p; };
  // persistent small buffers
  float* stats  = A(2*1024);
  float* enc    = A((size_t)BL*512);
  float* deg    = A((size_t)Nn);
  float* meanea = A((size_t)2*Nn);
  float* lmax   = A((size_t)Nn*2);
  float* wsum   = A((size_t)Nn*2);
  float* ghbuf  = A((size_t)BL*768);
  float* hcur   = A((size_t)BL*256);
  const size_t mark = off;
  // phase 1 (CNN/FC/GRU)
  float* act1  = A((size_t)S*64*180);
  float* act2  = A((size_t)S*64*180);
  float* pool1 = A((size_t)S*64*48);
  float* act3  = A((size_t)S*256*48);
  float* flat  = A((size_t)S*2048);
  float* fc1   = A((size_t)S*512);
  float* fc2   = A((size_t)S*256);
  float* gx    = A((size_t)S*768);
  float* h0    = A((size_t)S*256);
  float* h1    = A((size_t)S*256);
  // phase 2 (graph) overlaps phase 1 region (stream order makes this safe)
  off = mark;
  float* feat = A((size_t)Nn*513);
  float* xlb  = A((size_t)Nn*256);
  float* xrb  = A((size_t)Nn*256);
  float* g1   = A((size_t)Nn*256);
  float* g2   = A((size_t)Nn*128);
  float* g3   = A((size_t)Nn);
  float* elog = A((size_t)(E + Nn)*2);

  auto gemm = [&](const float* Am, const float* Bm, const float* bias, float* Cm,
                  int M, int N, int K, int transB, int relu){
    dim3 g((unsigned)cdiv(N,64), (unsigned)cdiv(M,64));
    wmma_gemm<<<g, dim3(128), 0, stream>>>(Am, Bm, bias, Cm, M, N, K, transB, relu);
  };
  auto conv = [&](const float* X, const float* Wt, const float* b, float* Y,
                  int IC, int OC, int D, int H, int W){
    int Ntot = S * D*H*W;
    dim3 g((unsigned)cdiv(Ntot,64), (unsigned)cdiv(OC,64));
    wmma_conv3d<<<g, dim3(128), 0, stream>>>(X, Wt, b, Y, S, IC, OC, D, H, W);
  };
  auto bn_nchw = [&](float* x, const float* g, const float* b, int C, int SP, int relu){
    bn_stats_nchw<<<dim3((unsigned)C), dim3(256), 0, stream>>>(x, stats, S, C, SP);
    bn_apply_nchw<<<dim3((unsigned)cdiv(S*C*SP,256)), dim3(256), 0, stream>>>(x, stats, g, b, S, C, SP, relu);
  };
  auto bn_rows = [&](float* x, const float* g, const float* b, int R, int C, int relu){
    bn_stats_rows<<<dim3((unsigned)C), dim3(256), 0, stream>>>(x, stats, R, C);
    bn_apply_rows<<<dim3((unsigned)cdiv(R*C,256)), dim3(256), 0, stream>>>(x, stats, g, b, R, C, relu);
  };
  auto fill = [&](float* p, float v, long long n){
    fill_f32<<<dim3((unsigned)((n + 255)/256)), dim3(256), 0, stream>>>(p, v, (int)n);
  };

  // ---------------------- Phase 1: CNN -> FC -> GRU -> dense ---------------
  conv(Xb,    P(7),  P(8),  act1, 5,  64, 5, 6, 6);          // conv1
  bn_nchw(act1, P(9),  P(10), 64, 180, 1);                   // bn1 + relu
  conv(act1,  P(11), P(12), act2, 64, 64, 5, 6, 6);          // conv2
  bn_nchw(act2, P(13), P(14), 64, 180, 1);                   // bn2 + relu
  maxpool3d_k<<<dim3((unsigned)cdiv(S*64*48,256)), dim3(256), 0, stream>>>(
      act2, pool1, S, 64, 5, 6, 6, 3, 4, 4, 1, 1, 1);
  conv(pool1, P(15), P(16), act3, 64, 256, 3, 4, 4);         // conv3
  bn_nchw(act3, P(17), P(18), 256, 48, 1);                   // bn3 + relu
  maxpool3d_k<<<dim3((unsigned)cdiv(S*256*8,256)), dim3(256), 0, stream>>>(
      act3, flat, S, 256, 3, 4, 4, 2, 2, 2, 1, 0, 0);        // -> [S,2048]

  gemm(flat, P(19), P(20), fc1, S, 512, 2048, 0, 0);         // lin1
  bn_rows(fc1, P(21), P(22), S, 512, 1);                     // bn4 + relu
  gemm(fc1,  P(23), P(24), fc2, S, 256, 512, 0, 0);          // lin2
  bn_rows(fc2, P(25), P(26), S, 256, 1);                     // bn5 + relu

  // GRU layer 0: gx = x @ wih^T + bih (wih is [768,256] -> transB)
  gemm(fc2, P(27), P(29), gx, S, 768, 256, 1, 0);
  fill(hcur, 0.f, (long long)BL*256);
  for (int t = 0; t < T; ++t){
    gemm(hcur, P(28), P(30), ghbuf, BL, 768, 256, 1, 0);     // gh = h @ whh^T + bhh
    gru_gate<<<dim3((unsigned)cdiv(BL*256,256)), dim3(256), 0, stream>>>(
        gx, ghbuf, hcur, h0, t, T, BL, 256);
  }
  // GRU layer 1
  gemm(h0, P(31), P(33), gx, S, 768, 256, 1, 0);
  fill(hcur, 0.f, (long long)BL*256);
  for (int t = 0; t < T; ++t){
    gemm(hcur, P(32), P(34), ghbuf, BL, 768, 256, 1, 0);
    gru_gate<<<dim3((unsigned)cdiv(BL*256,256)), dim3(256), 0, stream>>>(
        gx, ghbuf, hcur, h1, t, T, BL, 256);
  }
  // dense encoder: enc = relu(h1.reshape(36,6400) @ dense_w + b)
  gemm(h1, P(35), P(36), enc, BL, 512, 6400, 0, 1);

  // ---------------------- Phase 2: graph (GATv2 x3) ------------------------
  feat_gather<<<dim3((unsigned)cdiv(Nn*513,256)), dim3(256), 0, stream>>>(
      node_x, cell_idx, enc, feat, Nn);
  bn_rows(feat, P(37), P(38), Nn, 513, 0);                   // bng0, no relu

  fill(deg, 0.f, Nn);
  fill(meanea, 0.f, (long long)2*Nn);
  deg_meanea_accum<<<dim3((unsigned)cdiv(E,256)), dim3(256), 0, stream>>>(
      dst, edge_attr, deg, meanea, E);
  meanea_norm<<<dim3((unsigned)cdiv(Nn,256)), dim3(256), 0, stream>>>(meanea, deg, Nn);

  auto gatlayer = [&](const float* gin, int Fin, int Hh, int Cc,
                      const float* wl, const float* wr, const float* wep,
                      const float* attp, const float* bp, float* gout_){
    const int HC = Hh * Cc;
    const int E2 = E + Nn;
    gemm(gin, wl, nullptr, xlb, Nn, HC, Fin, 0, 0);
    gemm(gin, wr, nullptr, xrb, Nn, HC, Fin, 0, 0);
    fill(lmax, -INFINITY, (long long)Nn*Hh);
    fill(wsum, 0.f, (long long)Nn*Hh);
    fill(gout_, 0.f, (long long)Nn*HC);
    gat_edge_logits<<<dim3((unsigned)cdiv(E2,256)), dim3(256), 0, stream>>>(
        xlb, xrb, src, dst, edge_attr, meanea, wep, attp, elog, lmax, E, Nn, Hh, Cc);
    gat_edge_soft<<<dim3((unsigned)cdiv(E2*Hh,256)), dim3(256), 0, stream>>>(
        elog, lmax, wsum, dst, E, Nn, Hh);
    gat_edge_aggr<<<dim3((unsigned)cdiv(E2*HC,256)), dim3(256), 0, stream>>>(
        elog, wsum, xlb, dst, src, gout_, E, Nn, Hh, Cc);
    gat_finalize<<<dim3((unsigned)cdiv(Nn*HC,256)), dim3(256), 0, stream>>>(
        gout_, deg, bp, Nn, HC);
  };

  gatlayer(feat, 513, 2, 128, P(39), P(40), P(41), P(42), P(43), g1);
  bn_rows(g1, P(44), P(45), Nn, 256, 1);                     // bng1 + relu
  gatlayer(g1, 256, 1, 128, P(46), P(47), P(48), P(49), P(50), g2);
  bn_rows(g2, P(51), P(52), Nn, 128, 1);                     // bng2 + relu
  gatlayer(g2, 128, 1, 1,   P(53), P(54), P(55), P(56), P(57), g3);

  final_out<<<dim3((unsigned)cdiv(Nn,256)), dim3(256), 0, stream>>>(
      g3, train_mask, yv, (float*)d_out, Nn);
}